// MultiScaleRoIAlign_59038620451212
// MI455X (gfx1250) — compile-verified
//
#include <hip/hip_runtime.h>
#include <math.h>

typedef __attribute__((ext_vector_type(2))) float v2f;
typedef __attribute__((ext_vector_type(8))) float v8f;

#define OUTSZ 7
#define G     14      // OUT * sampling_ratio
#define NWAVE 4

// out = P * S * P^T per 16-channel block, P[7x16] pooling matrix (0.5 entries),
// contractions done with chained V_WMMA_F32_16X16X4_F32.
__global__ __launch_bounds__(128) void msroi_align_kernel(
    const float* __restrict__ f0, const float* __restrict__ f1,
    const float* __restrict__ f2, const float* __restrict__ f3,
    const float* __restrict__ boxes, float* __restrict__ out, int n_roi)
{
    // per-dimension bilinear tables: [0..G) = y, [G..2G) = x
    __shared__ int   s_i0[2 * G];
    __shared__ int   s_i1[2 * G];
    __shared__ float s_fr[2 * G];
    __shared__ float s_vd[2 * G];
    // stage-1 result T[wave][j][M][c]
    __shared__ float s_T[NWAVE][G][16][16];

    const int roi  = blockIdx.x;
    if (roi >= n_roi) return;
    const int b    = roi >> 8;          // N = 256 per batch
    const int n    = roi & 255;
    const int tid  = threadIdx.x;
    const int w    = tid >> 5;          // wave id (wave32)
    const int lane = tid & 31;
    const int half = lane >> 4;
    const int m    = lane & 15;

    // ---- ROI setup (uniform across block) ----
    const float* bx = boxes + ((size_t)((b << 8) + n)) * 4;
    const float x1 = bx[0], y1 = bx[1], x2 = bx[2], y2 = bx[3];

    // FPN LevelMapper: lvl = clip(floor(4 + log2(sqrt(area)/224) + 1e-6), 2, 5) - 2
    const float area = (x2 - x1) * (y2 - y1);
    const float sdim = sqrtf(fmaxf(area, 1e-30f));
    float lf = floorf(4.0f + log2f(sdim / 224.0f) + 1e-6f);
    lf = fminf(fmaxf(lf, 2.0f), 5.0f);
    const int lvl = (int)lf - 2;

    const float* feat; int L; float scale;
    if      (lvl == 0) { feat = f0; L = 200; scale = 0.25f;    }
    else if (lvl == 1) { feat = f1; L = 100; scale = 0.125f;   }
    else if (lvl == 2) { feat = f2; L = 50;  scale = 0.0625f;  }
    else               { feat = f3; L = 25;  scale = 0.03125f; }
    const int H = L, W = L;

    const float x1s = x1 * scale, y1s = y1 * scale;
    const float rw  = fmaxf(x2 * scale - x1s, 1.0f);
    const float rh  = fmaxf(y2 * scale - y1s, 1.0f);
    const float bw  = rw / (float)OUTSZ;
    const float bh  = rh / (float)OUTSZ;

    // ---- per-dimension sample tables ----
    if (tid < 2 * G) {
        const int  d     = tid / G;               // 0: y-dim, 1: x-dim
        const int  g     = tid % G;
        const float strt = d ? x1s : y1s;
        const float bs   = d ? bw  : bh;
        const int   Lim  = d ? W   : H;
        const float pos  = strt + (float)(g >> 1) * bs
                         + ((float)(g & 1) + 0.5f) * bs * 0.5f;
        const float valid = (pos >= -1.0f && pos <= (float)Lim) ? 1.0f : 0.0f;
        float pc = fminf(fmaxf(pos, 0.0f), (float)(Lim - 1));
        int   p0 = (int)floorf(pc);
        float fr = pc - (float)p0;
        int   p1 = p0 + 1; if (p1 > Lim - 1) p1 = Lim - 1;
        s_i0[tid] = p0; s_i1[tid] = p1; s_fr[tid] = fr; s_vd[tid] = valid;
    }
    __syncthreads();

    // ---- A operand: pooling matrix P (0.5 where k>>1 == m, m < 7) ----
    // 16x4 f32 A layout: lane = M; VGPR0 -> K = 4t + 2*half, VGPR1 -> K+1
    v2f a[4];
    for (int t = 0; t < 4; ++t) {
        const float pv = (m < OUTSZ && m == (2 * t + half)) ? 0.5f : 0.0f;
        a[t].x = pv; a[t].y = pv;
    }

    const size_t plane_sz = (size_t)H * W;

    for (int k = 0; k < 4; ++k) {
        const int c = (w * 4 + k) * 16 + m;       // this lane's channel
        const float* __restrict__ plane = feat + (size_t)((b << 8) + c) * plane_sz;

        // ---- stage 1: T[j][oy][c] = sum_i P[oy,i] * S[i][j][c] ----
        for (int j = 0; j < G; ++j) {
            const int   jx0 = s_i0[G + j], jx1 = s_i1[G + j];
            const float fx  = s_fr[G + j];
            const float vx  = s_vd[G + j];
            const float gx  = 1.0f - fx;

            // B operand (4x16 per step): VGPR0 -> K = 4t + 2*half, VGPR1 -> K+1
            v2f bv[4];
            for (int t = 0; t < 4; ++t) {
                float vals[2];
                for (int u = 0; u < 2; ++u) {
                    const int i  = 4 * t + 2 * half + u;
                    const int ic = (i < G) ? i : (G - 1);
                    const int   iy0 = s_i0[ic], iy1 = s_i1[ic];
                    const float fy  = s_fr[ic];
                    const float mk  = (i < G) ? (s_vd[ic] * vx) : 0.0f;
                    const float* r0 = plane + (size_t)iy0 * W;
                    const float* r1 = plane + (size_t)iy1 * W;
                    const float v00 = r0[jx0], v01 = r0[jx1];
                    const float v10 = r1[jx0], v11 = r1[jx1];
                    vals[u] = ((v00 * gx + v01 * fx) * (1.0f - fy)
                             + (v10 * gx + v11 * fx) * fy) * mk;
                }
                bv[t].x = vals[0]; bv[t].y = vals[1];
            }

            v8f acc = {};
            acc = __builtin_amdgcn_wmma_f32_16x16x4_f32(false, a[0], false, bv[0], (short)0, acc, false, false);
            acc = __builtin_amdgcn_wmma_f32_16x16x4_f32(false, a[1], false, bv[1], (short)0, acc, false, false);
            acc = __builtin_amdgcn_wmma_f32_16x16x4_f32(false, a[2], false, bv[2], (short)0, acc, false, false);
            acc = __builtin_amdgcn_wmma_f32_16x16x4_f32(false, a[3], false, bv[3], (short)0, acc, false, false);

            // D layout: VGPR r -> M = r + 8*half
            for (int r = 0; r < 8; ++r)
                s_T[w][j][r + 8 * half][m] = acc[r];
        }

        // ---- stage 2: out[ox][c] (per oy) = sum_j P[ox,j] * T[j][oy][c] ----
        for (int oy = 0; oy < OUTSZ; ++oy) {
            v2f bv[4];
            for (int t = 0; t < 4; ++t) {
                const int j0 = 4 * t + 2 * half;
                const int j1 = j0 + 1;
                bv[t].x = (j0 < G) ? s_T[w][j0][oy][m] : 0.0f;
                bv[t].y = (j1 < G) ? s_T[w][j1][oy][m] : 0.0f;
            }
            v8f acc = {};
            acc = __builtin_amdgcn_wmma_f32_16x16x4_f32(false, a[0], false, bv[0], (short)0, acc, false, false);
            acc = __builtin_amdgcn_wmma_f32_16x16x4_f32(false, a[1], false, bv[1], (short)0, acc, false, false);
            acc = __builtin_amdgcn_wmma_f32_16x16x4_f32(false, a[2], false, bv[2], (short)0, acc, false, false);
            acc = __builtin_amdgcn_wmma_f32_16x16x4_f32(false, a[3], false, bv[3], (short)0, acc, false, false);

            for (int r = 0; r < 8; ++r) {
                const int ox = r + 8 * half;
                if (ox < OUTSZ) {
                    out[(((size_t)roi * 256 + c) * OUTSZ + oy) * OUTSZ + ox] = acc[r];
                }
            }
        }
        // s_T is private to this wave; in-wave LDS ordering (DScnt) suffices
        // before the next k overwrites it.
    }
}

extern "C" void kernel_launch(void* const* d_in, const int* in_sizes, int n_in,
                              void* d_out, int out_size, void* d_ws, size_t ws_size,
                              hipStream_t stream) {
    const float* f0    = (const float*)d_in[0];
    const float* f1    = (const float*)d_in[1];
    const float* f2    = (const float*)d_in[2];
    const float* f3    = (const float*)d_in[3];
    const float* boxes = (const float*)d_in[4];
    float* out = (float*)d_out;

    const int n_roi = in_sizes[4] / 4;   // B*N = 512
    dim3 grid(n_roi), block(128);
    hipLaunchKernelGGL(msroi_align_kernel, grid, block, 0, stream,
                       f0, f1, f2, f3, boxes, out, n_roi);
}